// ScalarAffine_75831942578782
// MI455X (gfx1250) — compile-verified
//
#include <hip/hip_runtime.h>

// out = x * w + b  over 4*4096*2048 f32 elements (identity-GEMM reference is
// mathematically an elementwise affine). Pure HBM-bandwidth problem:
// 268 MB total traffic / 23.3 TB/s ~= 11.5 us floor. Optimal kernel is a
// 128-bit non-temporal stream; no matrix ops by design (0.25 FLOP/B is ~3
// orders of magnitude below the WMMA balance point).

typedef float v4f __attribute__((ext_vector_type(4)));

#define BLOCK  256   // 8 wave32 per block
#define UNROLL 4     // 4 x float4 = 64 B per thread

__global__ __launch_bounds__(BLOCK)
void scalar_affine_stream_kernel(const float* __restrict__ x,
                                 const float* __restrict__ weight,
                                 const float* __restrict__ bias,
                                 float* __restrict__ out,
                                 int nvec,   // number of float4 elements
                                 int n)      // total scalar elements
{
    const float w = weight[0];   // uniform scalar loads (s_load, cached)
    const float b = bias[0];

    const v4f* __restrict__ xv = (const v4f*)x;
    v4f* __restrict__ ov       = (v4f*)out;

    const int base = (int)blockIdx.x * (BLOCK * UNROLL) + (int)threadIdx.x;

    if (base + (UNROLL - 1) * BLOCK < nvec) {
        // Fast path (all blocks for the real shape): branch-free.
        // 4 back-to-back NT b128 loads -> one clause, all in flight;
        // then 16 v_fma; then 4 NT b128 stores.
        v4f v0 = __builtin_nontemporal_load(&xv[base + 0 * BLOCK]);
        v4f v1 = __builtin_nontemporal_load(&xv[base + 1 * BLOCK]);
        v4f v2 = __builtin_nontemporal_load(&xv[base + 2 * BLOCK]);
        v4f v3 = __builtin_nontemporal_load(&xv[base + 3 * BLOCK]);

        v4f r0 = v0 * w + b;
        v4f r1 = v1 * w + b;
        v4f r2 = v2 * w + b;
        v4f r3 = v3 * w + b;

        __builtin_nontemporal_store(r0, &ov[base + 0 * BLOCK]);
        __builtin_nontemporal_store(r1, &ov[base + 1 * BLOCK]);
        __builtin_nontemporal_store(r2, &ov[base + 2 * BLOCK]);
        __builtin_nontemporal_store(r3, &ov[base + 3 * BLOCK]);
    } else {
        // Ragged last block (never taken for 4*4096*2048): guarded per tile.
        for (int k = 0; k < UNROLL; ++k) {
            const int i = base + k * BLOCK;
            if (i < nvec) {
                v4f v = __builtin_nontemporal_load(&xv[i]);
                v4f r = v * w + b;
                __builtin_nontemporal_store(r, &ov[i]);
            }
        }
    }

    // Scalar tail (n % 4) — not hit for this shape; block 0 handles it.
    const int tail_base = nvec * 4;
    if (blockIdx.x == 0) {
        const int t = tail_base + (int)threadIdx.x;
        if (t < n) out[t] = fmaf(x[t], w, b);
    }
}

extern "C" void kernel_launch(void* const* d_in, const int* in_sizes, int n_in,
                              void* d_out, int out_size, void* d_ws, size_t ws_size,
                              hipStream_t stream)
{
    const float* x      = (const float*)d_in[0];
    const float* weight = (const float*)d_in[1];
    const float* bias   = (const float*)d_in[2];
    float* out          = (float*)d_out;

    const int n    = in_sizes[0];   // 33,554,432
    const int nvec = n >> 2;        // 8,388,608 float4s

    // Exact cover: 8,388,608 / (256*4) = 8192 blocks for this shape.
    const int per_block = BLOCK * UNROLL;
    int grid = (nvec + per_block - 1) / per_block;
    if (grid < 1) grid = 1;

    scalar_affine_stream_kernel<<<grid, BLOCK, 0, stream>>>(
        x, weight, bias, out, nvec, n);
}